// DenseDilatedKnnGraphDGL_3135326126138
// MI455X (gfx1250) — compile-verified
//
#include <hip/hip_runtime.h>

typedef float v2f __attribute__((ext_vector_type(2)));
typedef float v8f __attribute__((ext_vector_type(8)));

#define BATCH 16
#define NPTS  4096
#define CH    128
#define KNN   16
#define MAXD  3
#define LMAX  (KNN * MAXD)          // 48

#define WAVES_PER_WG   4
#define ROWS_PER_WAVE  32
#define ROWS_PER_WG    (WAVES_PER_WG * ROWS_PER_WAVE)   // 128
#define ROW_PAD        4
#define ROW_STRIDE     (CH + ROW_PAD)                   // 132 dwords -> conflict-free b64 reads
#define WGS_PER_BATCH  (NPTS / ROWS_PER_WG)             // 32
#define NTILES         (NPTS / 16)                      // 256

// low 32 bits of a generic pointer to LDS == LDS byte offset
__device__ __forceinline__ unsigned lds_off_u32(const void* p) {
    return (unsigned)(unsigned long long)p;
}

// CDNA5 async DMA: global -> LDS, 16 bytes, tracked by ASYNCcnt
__device__ __forceinline__ void async_copy_b128(const float* g, const void* lds) {
    unsigned l = lds_off_u32(lds);
    asm volatile("global_load_async_to_lds_b128 %0, %1, off"
                 :: "v"(l), "v"(g) : "memory");
}

// ---------------------------------------------------------------------------
// Kernel 1: per-point squared norms  sq[b*N+n] = sum_c x^2
// ---------------------------------------------------------------------------
__global__ __launch_bounds__(128) void sqnorm_kernel(const float* __restrict__ x,
                                                     float* __restrict__ sq) {
    int row = blockIdx.x * blockDim.x + threadIdx.x;          // 0 .. B*N-1
    const float4* p = (const float4*)(x + (size_t)row * CH);
    float s = 0.f;
#pragma unroll
    for (int i = 0; i < CH / 4; ++i) {
        float4 v = p[i];
        s += v.x * v.x + v.y * v.y + v.z * v.z + v.w * v.w;
    }
    sq[row] = s;
}

// ---------------------------------------------------------------------------
// Kernel 2: fused Gram (WMMA f32 16x16x4) + async-LDS candidate pipeline +
//           streaming top-L + dilated emit.
// One workgroup = 4 waves, each wave owns 32 query rows of one batch segment;
// candidate tiles are async-DMA'd to LDS once per WG and shared by all waves.
// ---------------------------------------------------------------------------
__global__ __launch_bounds__(128) void knn_dilated_kernel(const float* __restrict__ x,
                                                          const int* __restrict__ layer_idx_p,
                                                          const float* __restrict__ sq,
                                                          int* __restrict__ out_src,
                                                          int* __restrict__ out_dst) {
    __shared__ float xq[ROWS_PER_WG][ROW_STRIDE];                     // 67.5 KB
    __shared__ float xc[2][16][ROW_STRIDE];                           // 16.9 KB (double buffer)
    __shared__ float d2s[WAVES_PER_WG][ROWS_PER_WAVE][16];            //  8  KB
    __shared__ float listD[ROWS_PER_WG][LMAX];                        // 24  KB
    __shared__ int   listI[ROWS_PER_WG][LMAX];                        // 24  KB

    const int b     = blockIdx.x / WGS_PER_BATCH;
    const int rbase = (blockIdx.x % WGS_PER_BATCH) * ROWS_PER_WG;     // query row base in segment
    const float* xb  = x + (size_t)b * NPTS * CH;
    const float* sqb = sq + (size_t)b * NPTS;

    const int tid   = threadIdx.x;
    const int wave  = tid >> 5;
    const int lane  = tid & 31;
    const int nlane = lane & 15;           // column within 16-wide tile
    const int half  = lane >> 4;           // 0 or 1
    const int khalf = half * 2;            // k sub-offset for A/B fragments
    const int mlo   = wave * ROWS_PER_WAVE;

    // dilation from device-side layer_idx
    int dil = layer_idx_p[0] / 4 + 1;
    if (dil > MAXD) dil = MAXD;
    const int L = KNN * dil;               // 16 / 32 / 48

    // ---- prologue: async-stage the 128x128 query tile (coalesced b128 DMA) ----
    {
#pragma unroll
        for (int u = 0; u < 32; ++u) {
            int chunk = tid + u * 128;                 // 0..4095 16-byte chunks
            int row   = chunk >> 5;                    // 0..127
            int col   = (chunk & 31) * 4;              // 0..124 floats
            async_copy_b128(xb + (size_t)(rbase + row) * CH + col, &xq[row][col]);
        }
    }
    // ---- prologue: async-stage candidate tile 0 into buffer 0 ----
    {
#pragma unroll
        for (int u = 0; u < 4; ++u) {
            int chunk = tid + u * 128;                 // 0..511
            int row   = chunk >> 5;                    // 0..15
            int col   = (chunk & 31) * 4;
            async_copy_b128(xb + (size_t)row * CH + col, &xc[0][row][col]);
        }
    }
    asm volatile("s_wait_asynccnt 0x0" ::: "memory");
    __syncthreads();

    // loop-invariant query norms for this lane's accumulator rows
    float sqq0[8], sqq1[8];
#pragma unroll
    for (int v = 0; v < 8; ++v) {
        int mm = v + 8 * half;
        sqq0[v] = sqb[rbase + mlo + mm];
        sqq1[v] = sqb[rbase + mlo + 16 + mm];
    }

    // per-lane (= per query row) top-L state
    float* myD = listD[tid];
    int*   myI = listI[tid];
    int   cnt = 0, argmax = 0;
    float curmax = -3.0e38f;
    float* scr  = &d2s[wave][0][0];
    float* srow = &d2s[wave][lane][0];

    for (int nt = 0; nt < NTILES; ++nt) {
        const int cur = nt & 1;

        // prefetch next candidate tile into the other buffer (freed by the
        // end-of-previous-iteration barrier)
        if (nt + 1 < NTILES) {
#pragma unroll
            for (int u = 0; u < 4; ++u) {
                int chunk = tid + u * 128;
                int row   = chunk >> 5;
                int col   = (chunk & 31) * 4;
                async_copy_b128(xb + (size_t)((nt + 1) * 16 + row) * CH + col,
                                &xc[cur ^ 1][row][col]);
            }
        }

        // ---- Gram tile: 32 (rows) x 16 (cands), K=128 via 64 WMMAs ----
        v8f acc0 = {0.f, 0.f, 0.f, 0.f, 0.f, 0.f, 0.f, 0.f};
        v8f acc1 = {0.f, 0.f, 0.f, 0.f, 0.f, 0.f, 0.f, 0.f};
        const float* cbl = &xc[cur][nlane][0];

#pragma unroll
        for (int ks = 0; ks < CH / 4; ++ks) {
            const int kb = ks * 4 + khalf;
            v2f bf = *(const v2f*)(cbl + kb);                             // B frag (LDS)
            v2f a0 = *(const v2f*)(&xq[mlo + nlane][kb]);                 // A frag rows 0..15
            v2f a1 = *(const v2f*)(&xq[mlo + 16 + nlane][kb]);            // A frag rows 16..31
            acc0 = __builtin_amdgcn_wmma_f32_16x16x4_f32(false, a0, false, bf,
                                                         (short)0, acc0, false, false);
            acc1 = __builtin_amdgcn_wmma_f32_16x16x4_f32(false, a1, false, bf,
                                                         (short)0, acc1, false, false);
        }

        // d2 = |q|^2 + |c|^2 - 2*S ; spill 32x16 tile to wave-private scratch
        float sqc_n = sqb[nt * 16 + nlane];
#pragma unroll
        for (int v = 0; v < 8; ++v) {
            int mr = v + 8 * half;                                        // D-matrix row mapping
            scr[mr * 16 + nlane]        = __builtin_fmaf(-2.0f, acc0[v], sqq0[v] + sqc_n);
            scr[(16 + mr) * 16 + nlane] = __builtin_fmaf(-2.0f, acc1[v], sqq1[v] + sqc_n);
        }

        // wave-local LDS exchange: wait DS counter, block compiler reordering
        asm volatile("s_wait_dscnt 0" ::: "memory");
        __builtin_amdgcn_wave_barrier();

        // each lane scans its row's 16 new candidates into its top-L list
        const int idxbase = nt * 16;
#pragma unroll 1
        for (int j = 0; j < 16; ++j) {
            float d = srow[j];
            int   id = idxbase + j;
            if (cnt < L) {
                myD[cnt] = d;
                myI[cnt] = id;
                if (d > curmax) { curmax = d; argmax = cnt; }
                cnt++;
            } else if (d < curmax) {
                myD[argmax] = d;
                myI[argmax] = id;
                float m = -3.0e38f; int am = 0;
#pragma unroll 1
                for (int t = 0; t < L; ++t) {
                    float v = myD[t];
                    if (v > m) { m = v; am = t; }
                }
                curmax = m; argmax = am;
            }
        }

        // publish prefetched tile + free the buffer we will write next round
        asm volatile("s_wait_asynccnt 0x0" ::: "memory");
        __syncthreads();
    }

    // partial selection sort ascending (L <= 48), lane-private row
#pragma unroll 1
    for (int t = 0; t < L; ++t) {
        float best = myD[t]; int bi = t;
#pragma unroll 1
        for (int u = t + 1; u < L; ++u) {
            float v = myD[u];
            if (v < best) { best = v; bi = u; }
        }
        if (bi != t) {
            float td = myD[t]; myD[t] = myD[bi]; myD[bi] = td;
            int   ti = myI[t]; myI[t] = myI[bi]; myI[bi] = ti;
        }
    }

    // emit every dil-th neighbor: ranks 0, dil, 2*dil, ...
    const int gq = b * NPTS + rbase + tid;            // global query id
    int* so  = out_src + (size_t)gq * KNN;
    int* do_ = out_dst + (size_t)gq * KNN;
#pragma unroll
    for (int i = 0; i < KNN; ++i) {
        so[i]  = b * NPTS + myI[i * dil];
        do_[i] = gq;
    }
}

// ---------------------------------------------------------------------------
extern "C" void kernel_launch(void* const* d_in, const int* in_sizes, int n_in,
                              void* d_out, int out_size, void* d_ws, size_t ws_size,
                              hipStream_t stream) {
    (void)in_sizes; (void)n_in; (void)out_size; (void)ws_size;
    const float* x     = (const float*)d_in[0];
    const int*   layer = (const int*)d_in[1];
    float* sqn = (float*)d_ws;                         // B*N floats = 256 KB scratch
    int*   out = (int*)d_out;
    int*   out_src = out;
    int*   out_dst = out + (size_t)BATCH * NPTS * KNN;

    sqnorm_kernel<<<(BATCH * NPTS) / 128, 128, 0, stream>>>(x, sqn);
    knn_dilated_kernel<<<(BATCH * NPTS) / ROWS_PER_WG, 128, 0, stream>>>(
        x, layer, sqn, out_src, out_dst);
}